// KalmanGraphicalModel_35536559407574
// MI455X (gfx1250) — compile-verified
//
#include <hip/hip_runtime.h>

typedef __attribute__((ext_vector_type(2))) float v2f;
typedef __attribute__((ext_vector_type(8))) float v8f;

#define KGM_B 64
#define KGM_D 8
#define KGM_M 4
#define KGM_S 50000
#define TILES_PER_B (KGM_S / 16)          // 3125
#define TOTAL_TILES (KGM_B * TILES_PER_B) // 200000
#define CHUNK 8                           // s-tiles per wave (amortizes A loads)
#define GCOLS 28                          // K dim (8 xs + 8 past + 8 fut + 4 ys)

// ---------------------------------------------------------------------------
// Setup kernel: invert Q (8x8) and R (4x4) with Gauss-Jordan (SPD, well
// conditioned, no pivoting needed), then build the fused 32x28 matrix G:
//   rows  0..7 : [ -Qinv        | Qinv*F  | 0       | 0       ]
//   rows  8..15: [ -Ft*Qinv*F   | 0       | Ft*Qinv | 0       ]
//   rows 16..23: [ -Ht*Rinv*H   | 0       | 0       | Ht*Rinv ]
//   rows 24..31: zero padding (second 16-row WMMA tile well defined)
// ---------------------------------------------------------------------------
__global__ void kgm_setup(const float* __restrict__ F, const float* __restrict__ H,
                          const float* __restrict__ Q, const float* __restrict__ R,
                          float* __restrict__ G) {
  if (threadIdx.x != 0 || blockIdx.x != 0) return;

  // ---- invert Q ----
  float aq[8][16];
  for (int i = 0; i < 8; ++i)
    for (int j = 0; j < 8; ++j) {
      aq[i][j] = Q[i * 8 + j];
      aq[i][8 + j] = (i == j) ? 1.0f : 0.0f;
    }
  for (int p = 0; p < 8; ++p) {
    float inv = 1.0f / aq[p][p];
    for (int j = 0; j < 16; ++j) aq[p][j] *= inv;
    for (int r = 0; r < 8; ++r) {
      if (r == p) continue;
      float f = aq[r][p];
      for (int j = 0; j < 16; ++j) aq[r][j] -= f * aq[p][j];
    }
  }
  float Qi[8][8];
  for (int i = 0; i < 8; ++i)
    for (int j = 0; j < 8; ++j) Qi[i][j] = aq[i][8 + j];

  // ---- invert R ----
  float ar[4][8];
  for (int i = 0; i < 4; ++i)
    for (int j = 0; j < 4; ++j) {
      ar[i][j] = R[i * 4 + j];
      ar[i][4 + j] = (i == j) ? 1.0f : 0.0f;
    }
  for (int p = 0; p < 4; ++p) {
    float inv = 1.0f / ar[p][p];
    for (int j = 0; j < 8; ++j) ar[p][j] *= inv;
    for (int r = 0; r < 4; ++r) {
      if (r == p) continue;
      float f = ar[r][p];
      for (int j = 0; j < 8; ++j) ar[r][j] -= f * ar[p][j];
    }
  }
  float Ri[4][4];
  for (int i = 0; i < 4; ++i)
    for (int j = 0; j < 4; ++j) Ri[i][j] = ar[i][4 + j];

  // ---- derived products ----
  float QiF[8][8], FtQi[8][8], FtQiF[8][8], HtRi[8][4], HtRiH[8][8];
  for (int i = 0; i < 8; ++i)
    for (int j = 0; j < 8; ++j) {
      float s1 = 0.f, s2 = 0.f;
      for (int k = 0; k < 8; ++k) {
        s1 += Qi[i][k] * F[k * 8 + j];
        s2 += F[k * 8 + i] * Qi[k][j];
      }
      QiF[i][j] = s1;
      FtQi[i][j] = s2;
    }
  for (int i = 0; i < 8; ++i)
    for (int j = 0; j < 8; ++j) {
      float s = 0.f;
      for (int k = 0; k < 8; ++k) s += FtQi[i][k] * F[k * 8 + j];
      FtQiF[i][j] = s;
    }
  for (int i = 0; i < 8; ++i)
    for (int j = 0; j < 4; ++j) {
      float s = 0.f;
      for (int k = 0; k < 4; ++k) s += H[k * 8 + i] * Ri[k][j];
      HtRi[i][j] = s;
    }
  for (int i = 0; i < 8; ++i)
    for (int j = 0; j < 8; ++j) {
      float s = 0.f;
      for (int k = 0; k < 4; ++k) s += HtRi[i][k] * H[k * 8 + j];
      HtRiH[i][j] = s;
    }

  // ---- assemble G (32 x 28, row-major) ----
  for (int i = 0; i < 32 * GCOLS; ++i) G[i] = 0.0f;
  for (int i = 0; i < 8; ++i) {
    for (int j = 0; j < 8; ++j) {
      G[(i)      * GCOLS + j]      = -Qi[i][j];      // m1: -Qinv * xs
      G[(i)      * GCOLS + 8 + j]  =  QiF[i][j];     // m1: +Qinv F * x_past
      G[(8 + i)  * GCOLS + j]      = -FtQiF[i][j];   // m2: -FtQinvF * xs
      G[(8 + i)  * GCOLS + 16 + j] =  FtQi[i][j];    // m2: +FtQinv * x_fut
      G[(16 + i) * GCOLS + j]      = -HtRiH[i][j];   // m3: -HtRinvH * xs
    }
    for (int j = 0; j < 4; ++j)
      G[(16 + i) * GCOLS + 24 + j] = HtRi[i][j];     // m3: +HtRinv * ys
  }
}

// ---------------------------------------------------------------------------
// Main kernel: each wave32 owns CHUNK consecutive 16-column s-tiles.
// Per tile: OUT[32x16] = G[32x28] @ IN[28x16] via 14x V_WMMA_F32_16X16X4_F32.
// A fragments (constant G) are loaded once per wave and reused across tiles.
//
// A layout (16x4 f32, ISA 7.12.2): lanes 0-15 -> {K0,K1}, lanes 16-31 -> {K2,K3},
// M = lane&15. B layout (4x16): lanes 0-15 col N=lane rows {K0,K1}, lanes 16-31
// same cols rows {K2,K3}. C/D: VGPR g = row g (lanes 0-15) / row g+8 (16-31).
// ---------------------------------------------------------------------------
__global__ void __launch_bounds__(256) kgm_main(const float* __restrict__ xs,
                                                const float* __restrict__ ys,
                                                const float* __restrict__ G,
                                                float* __restrict__ out) {
  const int wave = blockIdx.x * (blockDim.x >> 5) + (threadIdx.x >> 5);
  if (wave * CHUNK >= TOTAL_TILES) return;  // wave-uniform (exact launch anyway)

  const int lane = threadIdx.x & 31;
  const int col  = lane & 15;
  const int half = lane >> 4;
  const int d0   = half * 2;  // rows (d0, d0+1) of each 4-row K block

  // ---- load the 2 A-tiles x 7 K-steps ONCE per wave (hits L2, 14x b64) ----
  v2f a[2][7];
#pragma unroll
  for (int m = 0; m < 2; ++m) {
    const float* gr = G + (m * 16 + col) * GCOLS;
#pragma unroll
    for (int k = 0; k < 7; ++k) {
      const int kc = k * 4 + half * 2;
      a[m][k].x = gr[kc];
      a[m][k].y = gr[kc + 1];
    }
  }

  const size_t plane = (size_t)KGM_B * KGM_D * KGM_S;  // elements per output
  float* m1 = out;
  float* m2 = out + plane;
  float* m3 = out + 2 * plane;

#pragma unroll 2
  for (int i = 0; i < CHUNK; ++i) {
    const int tile = wave * CHUNK + i;
    const int b  = tile / TILES_PER_B;
    const int t  = tile - b * TILES_PER_B;
    const int s  = t * 16 + col;                  // this lane's column (time)
    const int sp = (s > 0) ? s - 1 : 0;           // past (edge-replicated)
    const int sf = (s < KGM_S - 1) ? s + 1 : s;   // future (edge-replicated)

    const float* xb = xs + (size_t)b * KGM_D * KGM_S;
    const float* yb = ys + (size_t)b * KGM_S * KGM_M;

    // ---- gather the 7 B fragments for this lane (coalesced per half-wave) --
    v2f bv[7];
    bv[0].x = xb[(size_t)(d0 + 0) * KGM_S + s];   // IN rows 0..7  : xs[:, s]
    bv[0].y = xb[(size_t)(d0 + 1) * KGM_S + s];
    bv[1].x = xb[(size_t)(d0 + 4) * KGM_S + s];
    bv[1].y = xb[(size_t)(d0 + 5) * KGM_S + s];
    bv[2].x = xb[(size_t)(d0 + 0) * KGM_S + sp];  // IN rows 8..15 : xs[:, s-1]
    bv[2].y = xb[(size_t)(d0 + 1) * KGM_S + sp];
    bv[3].x = xb[(size_t)(d0 + 4) * KGM_S + sp];
    bv[3].y = xb[(size_t)(d0 + 5) * KGM_S + sp];
    bv[4].x = xb[(size_t)(d0 + 0) * KGM_S + sf];  // IN rows 16..23: xs[:, s+1]
    bv[4].y = xb[(size_t)(d0 + 1) * KGM_S + sf];
    bv[5].x = xb[(size_t)(d0 + 4) * KGM_S + sf];
    bv[5].y = xb[(size_t)(d0 + 5) * KGM_S + sf];
    bv[6].x = yb[(size_t)s * KGM_M + d0];         // IN rows 24..27: ys[s, :]
    bv[6].y = yb[(size_t)s * KGM_M + d0 + 1];

    // ---- 14 WMMAs: rows 0-15 (m1|m2) and rows 16-31 (m3|zero-pad) ----
    v8f acc0 = {};
    v8f acc1 = {};
#pragma unroll
    for (int k = 0; k < 7; ++k) {
      acc0 = __builtin_amdgcn_wmma_f32_16x16x4_f32(
          false, a[0][k], false, bv[k], (short)0, acc0, false, false);
      acc1 = __builtin_amdgcn_wmma_f32_16x16x4_f32(
          false, a[1][k], false, bv[k], (short)0, acc1, false, false);
    }

    // ---- scatter results (non-temporal: write-once stream, keep L2 for xs) --
    const size_t boff = (size_t)b * KGM_D * KGM_S;

    // acc0 vgpr g: row g (half==0 -> m1 row g) or row g+8 (half==1 -> m2 row g)
    float* p01 = (half ? m2 : m1) + boff + s;
#pragma unroll
    for (int g = 0; g < 8; ++g)
      __builtin_nontemporal_store(acc0[g], p01 + (size_t)g * KGM_S);

    // acc1 vgpr g: row 16+g (half==0 -> m3 row g); rows 24-31 are zero padding
    if (half == 0) {
      float* p3 = m3 + boff + s;
#pragma unroll
      for (int g = 0; g < 8; ++g)
        __builtin_nontemporal_store(acc1[g], p3 + (size_t)g * KGM_S);
    }
  }
}

extern "C" void kernel_launch(void* const* d_in, const int* in_sizes, int n_in,
                              void* d_out, int out_size, void* d_ws, size_t ws_size,
                              hipStream_t stream) {
  const float* xs = (const float*)d_in[0];
  const float* ys = (const float*)d_in[1];
  const float* F  = (const float*)d_in[2];
  const float* H  = (const float*)d_in[3];
  const float* Q  = (const float*)d_in[4];
  const float* R  = (const float*)d_in[5];
  float* G = (float*)d_ws;  // 32*28 floats = 3584 B of scratch

  kgm_setup<<<1, 1, 0, stream>>>(F, H, Q, R, G);

  // 200000 tiles / 8 per wave = 25000 waves; 8 waves per 256-thread block.
  kgm_main<<<dim3(TOTAL_TILES / CHUNK / 8), dim3(256), 0, stream>>>(
      xs, ys, G, (float*)d_out);
}